// AttentionOneHotConv_56839597195349
// MI455X (gfx1250) — compile-verified
//
#include <hip/hip_runtime.h>
#include <hip/hip_bf16.h>

// ---------------------------------------------------------------------------
// Problem constants (from reference)
// ---------------------------------------------------------------------------
#define GNUM   64      // graphs
#define NPG    256     // nodes per graph
#define EPG    4096    // edges per graph
#define NTOT   (GNUM * NPG)   // 16384 nodes
#define HH     4       // heads
#define CC     64      // channels per head
#define HC     (HH * CC)      // 256
#define OHC    8
#define KDIM   264     // IN + OHC
#define ASTR   265     // LDS row stride for A staging (odd -> conflict-free)

typedef float v2f __attribute__((ext_vector_type(2)));
typedef float v8f __attribute__((ext_vector_type(8)));

// f32 WMMA: D(16x16) = A(16x4) * B(4x16) + C, exact fp32 (matches reference math)
__device__ inline v8f wmma_f32_4(v2f a, v2f b, v8f c) {
#if __has_builtin(__builtin_amdgcn_wmma_f32_16x16x4_f32)
  return __builtin_amdgcn_wmma_f32_16x16x4_f32(false, a, false, b, (short)0, c,
                                               false, false);
#else
  c[0] += a.x * b.x + a.y * b.y;   // host-pass / fallback placeholder
  return c;
#endif
}

// ---------------------------------------------------------------------------
// Kernel 1: per-node sort(256) -> conv1d(1->8,k3) -> relu -> conv1d(8->16,k3)
//           -> relu -> mean(L) -> fc(16->8).  One 256-thread block per node.
// ---------------------------------------------------------------------------
__global__ void k1_sortconv(const float* __restrict__ onehots,
                            const float* __restrict__ w1, const float* __restrict__ b1,
                            const float* __restrict__ w2, const float* __restrict__ b2,
                            const float* __restrict__ fcw, const float* __restrict__ fcb,
                            float* __restrict__ ohfeat) {
  __shared__ float s[NPG];
  __shared__ float y1[8 * 258];   // 8 channels, L=256 with halo
  __shared__ float zacc[16];
  const int t = threadIdx.x;
  const int n = blockIdx.x;

  s[t] = onehots[(size_t)n * NPG + t];
  if (t < 16) zacc[t] = 0.f;
  __syncthreads();

  // Bitonic sort ascending, 256 elements, one per thread.
  for (int k = 2; k <= NPG; k <<= 1) {
    for (int j = k >> 1; j > 0; j >>= 1) {
      int ixj = t ^ j;
      if (ixj > t) {
        float a = s[t], b = s[ixj];
        bool up = ((t & k) == 0);
        if ((a > b) == up) { s[t] = b; s[ixj] = a; }
      }
      __syncthreads();
    }
  }

  // conv1 (1->8), zero 'same' padding
  {
    float lft = (t > 0)       ? s[t - 1] : 0.f;
    float mid = s[t];
    float rgt = (t < NPG - 1) ? s[t + 1] : 0.f;
#pragma unroll
    for (int c = 0; c < 8; ++c) {
      float v = b1[c] + w1[c * 3 + 0] * lft + w1[c * 3 + 1] * mid + w1[c * 3 + 2] * rgt;
      y1[c * 258 + t + 1] = fmaxf(v, 0.f);
    }
  }
  if (t < 8) { y1[t * 258 + 0] = 0.f; y1[t * 258 + 257] = 0.f; }
  __syncthreads();

  // conv2 (8->16) + relu + accumulate mean over L
  float part[16];
#pragma unroll
  for (int c2 = 0; c2 < 16; ++c2) part[c2] = b2[c2];
#pragma unroll
  for (int c = 0; c < 8; ++c) {
    float a0 = y1[c * 258 + t + 0];
    float a1 = y1[c * 258 + t + 1];
    float a2 = y1[c * 258 + t + 2];
#pragma unroll
    for (int c2 = 0; c2 < 16; ++c2) {
      const float* w = &w2[(c2 * 8 + c) * 3];
      part[c2] += w[0] * a0 + w[1] * a1 + w[2] * a2;
    }
  }
#pragma unroll
  for (int c2 = 0; c2 < 16; ++c2) atomicAdd(&zacc[c2], fmaxf(part[c2], 0.f));
  __syncthreads();

  if (t < OHC) {
    float acc = fcb[t];
#pragma unroll
    for (int c2 = 0; c2 < 16; ++c2)
      acc += (zacc[c2] * (1.f / (float)NPG)) * fcw[c2 * OHC + t];
    ohfeat[(size_t)n * OHC + t] = acc;
  }
}

// ---------------------------------------------------------------------------
// Kernel 2: projection GEMM xp[N,256] = [x | oh_feat][N,264] @ lin_w[264,256]
// via V_WMMA_F32_16X16X4_F32.
// One block (8 waves) per 16-row stripe.  The [16 x 264] A block is staged
// once into LDS (shared by all 8 waves); each wave computes TWO 16x16 column
// tiles (tn = w and w+8) with independent accumulator chains.
// ---------------------------------------------------------------------------
__global__ void k2_proj_wmma(const float* __restrict__ x,
                             const float* __restrict__ ohfeat,
                             const float* __restrict__ linw,
                             float* __restrict__ xp) {
  __shared__ float As[16 * ASTR];           // 16 rows x 264 (stride 265)
  const int t    = threadIdx.x;
  const int wave = t >> 5;
  const int lane = t & 31;
  const int tm   = blockIdx.x;              // 0..1023 row stripe
  const int m    = lane & 15;
  const int koff = (lane >> 4) << 1;        // 0 or 2
  const int half = lane >> 4;

  // Stage A = [x | oh_feat] rows tm*16 .. tm*16+15 into LDS.
  for (int i = t; i < 16 * 256; i += 256) {
    int r = i >> 8, c = i & 255;
    As[r * ASTR + c] = x[(size_t)(tm * 16 + r) * 256 + c];
  }
  if (t < 16 * OHC) {
    int r = t >> 3, c = t & 7;
    As[r * ASTR + 256 + c] = ohfeat[(size_t)(tm * 16 + r) * OHC + c];
  }
  __syncthreads();

  const int col0 = wave * 16 + m;           // tn0 = wave
  const int col1 = (wave + 8) * 16 + m;     // tn1 = wave + 8
  const float* arow = &As[m * ASTR];

  v8f acc0 = {}, acc1 = {};
  for (int k0 = 0; k0 < KDIM; k0 += 4) {
    __builtin_prefetch(linw + (size_t)(k0 + 8) * HC + col0, 0, 1);
    v2f a, b0, b1;
    a.x  = arow[k0 + koff];
    a.y  = arow[k0 + koff + 1];
    b0.x = linw[(size_t)(k0 + koff) * HC + col0];
    b0.y = linw[(size_t)(k0 + koff + 1) * HC + col0];
    b1.x = linw[(size_t)(k0 + koff) * HC + col1];
    b1.y = linw[(size_t)(k0 + koff + 1) * HC + col1];
    acc0 = wmma_f32_4(a, b0, acc0);
    acc1 = wmma_f32_4(a, b1, acc1);
  }
  // D layout: vgpr r -> M = r + 8*half, N = lane&15
#pragma unroll
  for (int r = 0; r < 8; ++r) {
    size_t orow = (size_t)(tm * 16 + r + half * 8) * HC;
    xp[orow + col0] = acc0[r];
    xp[orow + col1] = acc1[r];
  }
}

// ---------------------------------------------------------------------------
// Kernel 2b: attention logits  alpha_l/r[n,h] = sum_c xp[n,h,c]*att_{l,r}[h,c]
// ---------------------------------------------------------------------------
__global__ void k2b_alpha(const float* __restrict__ xp,
                          const float* __restrict__ attl,
                          const float* __restrict__ attr,
                          float* __restrict__ al, float* __restrict__ ar) {
  int idx = blockIdx.x * blockDim.x + threadIdx.x;   // n*H + h
  if (idx >= NTOT * HH) return;
  int h = idx & (HH - 1);
  int n = idx >> 2;
  const float* xr = xp + (size_t)n * HC + h * CC;
  float sl = 0.f, sr = 0.f;
#pragma unroll 8
  for (int c = 0; c < CC; ++c) {
    float v = xr[c];
    sl += v * attl[h * CC + c];
    sr += v * attr[h * CC + c];
  }
  al[idx] = sl;
  ar[idx] = sr;
}

// ---------------------------------------------------------------------------
// Kernel 3: per-graph edge softmax + weighted scatter-add.
// One 1024-thread block (32 wave32s) per graph for gather-latency hiding.
// The [256 x 256] fp32 accumulator (256KB) lives in CDNA5's 320KB LDS ->
// all 1M per-graph edge accumulations are LDS atomics; xp[src] rows stream
// from the 192MB L2 (xp is 16MB, fully resident).
// Dynamic LDS: agg 65536 | al 1024 | ar 1024 | mx 1024 | sden 1024 floats.
// ---------------------------------------------------------------------------
__global__ void k3_attn(const float* __restrict__ xp,
                        const int* __restrict__ adjs,
                        const float* __restrict__ alg,
                        const float* __restrict__ arg2,
                        const float* __restrict__ bias,
                        float* __restrict__ xout) {
  extern __shared__ float sm[];
  float* agg  = sm;                 // 65536
  float* al   = sm + 65536;         // 1024  (node*4 + h)
  float* ar   = al + 1024;          // 1024
  float* mx   = ar + 1024;          // 1024
  float* sden = mx + 1024;          // 1024
  const int t = threadIdx.x;        // 0..1023
  const int g = blockIdx.x;
  const int* srcs = adjs + (size_t)g * 2 * EPG;
  const int* dsts = srcs + EPG;

  for (int i = t; i < NPG * HC; i += 1024) agg[i] = 0.f;
  // al/ar are flat [256*4]: one element per thread
  al[t]   = alg[(size_t)g * NPG * HH + t];
  ar[t]   = arg2[(size_t)g * NPG * HH + t];
  mx[t]   = -1e30f;
  sden[t] = 0.f;
  __syncthreads();

  // pass 1: segment max over receivers (CAS float-max in LDS)
  for (int e = t; e < EPG; e += 1024) {
    int s = srcs[e], d = dsts[e];
#pragma unroll
    for (int h = 0; h < HH; ++h) {
      float el = al[s * HH + h] + ar[d * HH + h];
      el = (el > 0.f) ? el : 0.2f * el;
      int* addr = (int*)&mx[d * HH + h];
      int old = *addr;
      while (__int_as_float(old) < el) {
        int assumed = old;
        old = atomicCAS(addr, assumed, __float_as_int(el));
        if (old == assumed) break;
      }
    }
  }
  __syncthreads();

  // pass 2: segment sum of exp
  for (int e = t; e < EPG; e += 1024) {
    int s = srcs[e], d = dsts[e];
#pragma unroll
    for (int h = 0; h < HH; ++h) {
      float el = al[s * HH + h] + ar[d * HH + h];
      el = (el > 0.f) ? el : 0.2f * el;
      atomicAdd(&sden[d * HH + h], __expf(el - mx[d * HH + h]));
    }
  }
  __syncthreads();

  // pass 3: weighted scatter.  One wave per edge; lane l covers
  // h = l/8, channels c0..c0+7 (8 lanes/head * 8 ch = 64 per head).
  {
    const int wave = t >> 5, lane = t & 31;   // 32 waves
    const int h  = lane >> 3;
    const int c0 = (lane & 7) << 3;
    for (int e = wave; e < EPG; e += 32) {
      int s = srcs[e], d = dsts[e];
      float el = al[s * HH + h] + ar[d * HH + h];
      el = (el > 0.f) ? el : 0.2f * el;
      float a = __expf(el - mx[d * HH + h]) / (sden[d * HH + h] + 1e-16f);
      const float4* xr4 =
          (const float4*)(xp + ((size_t)g * NPG + s) * HC + h * CC + c0);
      float4 p0 = xr4[0];
      float4 p1 = xr4[1];
      float* ag = agg + d * HC + h * CC + c0;
      atomicAdd(&ag[0], p0.x * a);
      atomicAdd(&ag[1], p0.y * a);
      atomicAdd(&ag[2], p0.z * a);
      atomicAdd(&ag[3], p0.w * a);
      atomicAdd(&ag[4], p1.x * a);
      atomicAdd(&ag[5], p1.y * a);
      atomicAdd(&ag[6], p1.z * a);
      atomicAdd(&ag[7], p1.w * a);
    }
  }
  __syncthreads();

  // write out with bias
  for (int i = t; i < NPG * HC; i += 1024)
    xout[(size_t)g * NPG * HC + i] = agg[i] + bias[i & (HC - 1)];
}

// ---------------------------------------------------------------------------
// Kernel 4: per-graph one-hot aggregation as a dense GEMM:
//   new_oh = (Count + I) @ oh,  Count[d][s] = #edges (s->d).
// Count (256KB) built in LDS with cheap atomics, then WMMA f32 16x16x4 GEMM.
// Each wave computes two column tiles per row stripe (dual WMMA chains,
// shared A reads).  Replaces 67M global scatter atomics with matrix work.
// ---------------------------------------------------------------------------
__global__ void k4_ohagg_wmma(const float* __restrict__ oh,
                              const int* __restrict__ adjs,
                              float* __restrict__ newoh) {
  extern __shared__ float cnt[];    // 65536 floats = 256KB
  const int t = threadIdx.x;
  const int g = blockIdx.x;
  const int* srcs = adjs + (size_t)g * 2 * EPG;
  const int* dsts = srcs + EPG;

  for (int i = t; i < NPG * NPG; i += 256) cnt[i] = 0.f;
  __syncthreads();
  for (int e = t; e < EPG; e += 256)
    atomicAdd(&cnt[dsts[e] * NPG + srcs[e]], 1.f);
  __syncthreads();
  cnt[t * NPG + t] += 1.f;          // + identity (residual +oh_g)
  __syncthreads();

  const int wave = t >> 5, lane = t & 31;
  const int m    = lane & 15;
  const int koff = (lane >> 4) << 1;
  const int half = lane >> 4;
  const float* ohg  = oh    + (size_t)g * NPG * NPG;
  float*       outg = newoh + (size_t)g * NPG * NPG;

  const int col0 = wave * 16 + m;           // tn0 = wave
  const int col1 = (wave + 8) * 16 + m;     // tn1 = wave + 8

  for (int tm = 0; tm < 16; ++tm) {         // 16 row stripes
    const float* arow = &cnt[(tm * 16 + m) * NPG];
    v8f acc0 = {}, acc1 = {};
    for (int k0 = 0; k0 < NPG; k0 += 4) {
      v2f a, b0, b1;
      a.x  = arow[k0 + koff];
      a.y  = arow[k0 + koff + 1];
      b0.x = ohg[(size_t)(k0 + koff) * NPG + col0];
      b0.y = ohg[(size_t)(k0 + koff + 1) * NPG + col0];
      b1.x = ohg[(size_t)(k0 + koff) * NPG + col1];
      b1.y = ohg[(size_t)(k0 + koff + 1) * NPG + col1];
      acc0 = wmma_f32_4(a, b0, acc0);
      acc1 = wmma_f32_4(a, b1, acc1);
    }
#pragma unroll
    for (int r = 0; r < 8; ++r) {
      size_t orow = (size_t)(tm * 16 + r + half * 8) * NPG;
      outg[orow + col0] = acc0[r];
      outg[orow + col1] = acc1[r];
    }
  }
}

// ---------------------------------------------------------------------------
// Launch
// ---------------------------------------------------------------------------
extern "C" void kernel_launch(void* const* d_in, const int* in_sizes, int n_in,
                              void* d_out, int out_size, void* d_ws, size_t ws_size,
                              hipStream_t stream) {
  const float* x       = (const float*)d_in[0];
  const float* onehots = (const float*)d_in[1];
  const int*   adjs    = (const int*)d_in[2];
  const float* lin_w   = (const float*)d_in[3];
  const float* att_l   = (const float*)d_in[4];
  const float* att_r   = (const float*)d_in[5];
  const float* bias    = (const float*)d_in[6];
  const float* c1w     = (const float*)d_in[7];
  const float* c1b     = (const float*)d_in[8];
  const float* c2w     = (const float*)d_in[9];
  const float* c2b     = (const float*)d_in[10];
  const float* fcw     = (const float*)d_in[11];
  const float* fcb     = (const float*)d_in[12];

  float* out_x  = (float*)d_out;                       // [N, 256]
  float* out_oh = out_x + (size_t)NTOT * HC;           // [G, 256, 256]

  // workspace carve (~17.8 MB)
  float* xp  = (float*)d_ws;                           // [N, 256]
  float* ohf = xp  + (size_t)NTOT * HC;                // [N, 8]
  float* al  = ohf + (size_t)NTOT * OHC;               // [N, 4]
  float* ar  = al  + (size_t)NTOT * HH;                // [N, 4]

  // 1) sort + conv pipe -> oh_feat
  k1_sortconv<<<NTOT, 256, 0, stream>>>(onehots, c1w, c1b, c2w, c2b, fcw, fcb, ohf);

  // 2) projection GEMM (WMMA f32): one block per 16-row stripe
  k2_proj_wmma<<<NTOT / 16, 256, 0, stream>>>(x, ohf, lin_w, xp);

  // 2b) attention logits
  k2b_alpha<<<(NTOT * HH + 255) / 256, 256, 0, stream>>>(xp, att_l, att_r, al, ar);

  // 3) per-graph edge softmax + LDS scatter (272KB dynamic LDS, 32 waves)
  k3_attn<<<GNUM, 1024, (65536 + 4 * 1024) * sizeof(float), stream>>>(
      xp, adjs, al, ar, bias, out_x);

  // 4) per-graph one-hot aggregation GEMM (256KB dynamic LDS)
  k4_ohagg_wmma<<<GNUM, 256, 65536 * sizeof(float), stream>>>(onehots, adjs, out_oh);
}